// splineconv_69801808494889
// MI455X (gfx1250) — compile-verified
//
#include <hip/hip_runtime.h>
#include <math.h>

// Fixed problem sizes from the reference.
constexpr int NN   = 20000;   // nodes (20000 = 1250 * 16 -> exact WMMA tiling)
constexpr int EE   = 640000;  // edges
constexpr int F_IN = 128;
constexpr int F_H  = 64;
constexpr int F_O  = 16;
constexpr int KT   = 4;       // spline taps

typedef __attribute__((ext_vector_type(2))) float v2f;
typedef __attribute__((ext_vector_type(8))) float v8f;

// ---------------------------------------------------------------------------
// WMMA f32 GEMM: C[M x (16*tiles_n)] = A[M x Kd] * B[Kd x (16*tiles_n)] (+bias)
// One 16x16 C tile per wave using V_WMMA_F32_16X16X4_F32, K-step = 4.
// Fragment layouts per CDNA5 ISA 7.12.2 (32-bit A 16x4, 32-bit C/D 16x16).
// All tiles are full (M multiple of 16), EXEC is all-ones as WMMA requires.
// ---------------------------------------------------------------------------
__global__ void wmma_gemm_f32(const float* __restrict__ A, int lda,
                              const float* __restrict__ B, int ldb,
                              const float* __restrict__ bias,
                              float* __restrict__ C, int ldc,
                              int Kd, int tiles_n, int tiles_total)
{
    const int wave = (blockIdx.x * blockDim.x + threadIdx.x) >> 5;
    const int lane = threadIdx.x & 31;
    if (wave >= tiles_total) return;           // uniform per-wave exit

    const int tm = wave / tiles_n;
    const int tn = wave - tm * tiles_n;

    const int row  = tm * 16 + (lane & 15);    // A row this lane feeds
    const int col  = tn * 16 + (lane & 15);    // B/C column this lane holds
    const int koff = (lane >> 4) << 1;         // lanes 0-15: K+{0,1}; 16-31: K+{2,3}

    v8f acc = {};
    for (int kk = 0; kk < Kd; kk += 4) {
        v2f a, b;
        a.x = A[row * lda + kk + koff];        // contiguous pair -> b64 load
        a.y = A[row * lda + kk + koff + 1];
        b.x = B[(kk + koff) * ldb + col];
        b.y = B[(kk + koff + 1) * ldb + col];
        acc = __builtin_amdgcn_wmma_f32_16x16x4_f32(
                  false, a, false, b, (short)0, acc, false, false);
    }

    const float badd  = bias ? bias[col] : 0.0f;
    const int   rbase = tm * 16 + ((lane >> 4) << 3);  // upper half-wave: rows +8
#pragma unroll
    for (int v = 0; v < 8; ++v)
        C[(rbase + v) * ldc + col] = acc[v] + badd;
}

// ---------------------------------------------------------------------------
// Per-edge spline blend + scatter-add. Each thread owns one float4 chunk of
// one edge's feature row (F/4 chunks per edge, F = 64 or 16). The whole xw
// table (<= 20.5 MB) is L2-resident on MI455X (192 MB L2), so the random
// src-gathers and dst-atomics never touch HBM.
// ---------------------------------------------------------------------------
__global__ void edge_scatter(const float* __restrict__ xw,     // [KT][NN][F]
                             const int*   __restrict__ eidx,   // [2][EE]
                             const float* __restrict__ eattr,  // [EE]
                             float* __restrict__ agg,          // [NN][F]
                             float* __restrict__ cnt,          // [NN] or null
                             int F, int shift)                 // shift = log2(F/4)
{
    const int t = blockIdx.x * blockDim.x + threadIdx.x;
    const int total = EE << shift;
    if (t >= total) return;
    const int e = t >> shift;
    const int q = t & ((1 << shift) - 1);

    // degree-1 open B-spline: two taps, weights (1-frac, frac)
    float pos = eattr[e] * (float)(KT - 1);
    float k0f = floorf(pos);
    k0f = fminf(fmaxf(k0f, 0.0f), (float)(KT - 1));
    const float frac = pos - k0f;
    const int k0 = (int)k0f;
    const int k1 = min(k0 + 1, KT - 1);

    const int src = eidx[e];
    const int dst = eidx[EE + e];

    const float4 a = *((const float4*)(xw + (size_t)k0 * NN * F + (size_t)src * F) + q);
    const float4 b = *((const float4*)(xw + (size_t)k1 * NN * F + (size_t)src * F) + q);
    const float w0 = 1.0f - frac, w1 = frac;

    float* ap = agg + (size_t)dst * F + (q << 2);
    atomicAdd(ap + 0, w0 * a.x + w1 * b.x);
    atomicAdd(ap + 1, w0 * a.y + w1 * b.y);
    atomicAdd(ap + 2, w0 * a.z + w1 * b.z);
    atomicAdd(ap + 3, w0 * a.w + w1 * b.w);

    if (cnt != nullptr && q == 0) atomicAdd(cnt + dst, 1.0f);  // in-degree
}

__global__ void zero_f32(float* __restrict__ p, int n)
{
    const int t = blockIdx.x * blockDim.x + threadIdx.x;
    if (t < n) p[t] = 0.0f;
}

// h = relu(agg/max(cnt,1) + rootout)   (rootout already includes bias)
__global__ void finalize_relu(const float* __restrict__ agg,
                              const float* __restrict__ cnt,
                              const float* __restrict__ rootout,
                              float* __restrict__ h, int total, int fshift)
{
    const int t = blockIdx.x * blockDim.x + threadIdx.x;
    if (t >= total) return;
    const int n = t >> fshift;
    const float c = fmaxf(cnt[n], 1.0f);
    h[t] = fmaxf(agg[t] / c + rootout[t], 0.0f);
}

// out = log_softmax(relu(agg/max(cnt,1) + rootout)) over F_O=16 columns
__global__ void finalize_logsoftmax(const float* __restrict__ agg,
                                    const float* __restrict__ cnt,
                                    const float* __restrict__ rootout,
                                    float* __restrict__ out)
{
    const int n = blockIdx.x * blockDim.x + threadIdx.x;
    if (n >= NN) return;
    const float inv = 1.0f / fmaxf(cnt[n], 1.0f);
    float v[F_O];
    float m = -INFINITY;
#pragma unroll
    for (int i = 0; i < F_O; ++i) {
        v[i] = fmaxf(agg[n * F_O + i] * inv + rootout[n * F_O + i], 0.0f);
        m = fmaxf(m, v[i]);
    }
    float s = 0.0f;
#pragma unroll
    for (int i = 0; i < F_O; ++i) s += expf(v[i] - m);
    const float ls = logf(s);
#pragma unroll
    for (int i = 0; i < F_O; ++i) out[n * F_O + i] = v[i] - m - ls;
}

extern "C" void kernel_launch(void* const* d_in, const int* in_sizes, int n_in,
                              void* d_out, int out_size, void* d_ws, size_t ws_size,
                              hipStream_t stream)
{
    const float* x     = (const float*)d_in[0];
    const int*   eidx  = (const int*)  d_in[1];
    const float* eattr = (const float*)d_in[2];
    const float* W1    = (const float*)d_in[3];
    const float* root1 = (const float*)d_in[4];
    const float* b1    = (const float*)d_in[5];
    const float* W2    = (const float*)d_in[6];
    const float* root2 = (const float*)d_in[7];
    const float* b2    = (const float*)d_in[8];
    float* out = (float*)d_out;

    // Workspace layout (all offsets multiples of 4 floats -> float4-aligned).
    float* ws   = (float*)d_ws;
    float* xw1  = ws;                                   // KT*NN*F_H = 5,120,000
    float* ro1  = xw1 + (size_t)KT * NN * F_H;          // NN*F_H
    float* agg1 = ro1 + (size_t)NN * F_H;               // NN*F_H
    float* cnt  = agg1 + (size_t)NN * F_H;              // NN   (contiguous w/ agg1)
    float* h    = cnt + NN;                             // NN*F_H
    float* xw2  = h + (size_t)NN * F_H;                 // KT*NN*F_O
    float* ro2  = xw2 + (size_t)KT * NN * F_O;          // NN*F_O
    float* agg2 = ro2 + (size_t)NN * F_O;               // NN*F_O

    const int BS = 256;

    // Zero the accumulators (agg1 and cnt are contiguous).
    {
        int n1 = NN * F_H + NN;
        zero_f32<<<(n1 + BS - 1) / BS, BS, 0, stream>>>(agg1, n1);
        int n2 = NN * F_O;
        zero_f32<<<(n2 + BS - 1) / BS, BS, 0, stream>>>(agg2, n2);
    }

    // ---- Layer 1: xw1[k] = x @ W1[k]; ro1 = x @ root1 + b1 ------------------
    {
        const int tiles_n = F_H / 16;                   // 4
        const int tiles_total = (NN / 16) * tiles_n;    // 5000
        const int grid = (tiles_total + 7) / 8;         // 8 waves / block
        for (int k = 0; k < KT; ++k)
            wmma_gemm_f32<<<grid, BS, 0, stream>>>(
                x, F_IN, W1 + (size_t)k * F_IN * F_H, F_H, nullptr,
                xw1 + (size_t)k * NN * F_H, F_H, F_IN, tiles_n, tiles_total);
        wmma_gemm_f32<<<grid, BS, 0, stream>>>(
            x, F_IN, root1, F_H, b1, ro1, F_H, F_IN, tiles_n, tiles_total);
    }

    // Edge blend + scatter (F_H=64 -> 16 float4 chunks/edge), builds cnt.
    {
        const int total = EE << 4;
        edge_scatter<<<(total + BS - 1) / BS, BS, 0, stream>>>(
            xw1, eidx, eattr, agg1, cnt, F_H, 4);
    }

    // h = relu(mean + root)
    {
        const int total = NN * F_H;
        finalize_relu<<<(total + BS - 1) / BS, BS, 0, stream>>>(
            agg1, cnt, ro1, h, total, 6);
    }

    // ---- Layer 2: xw2[k] = h @ W2[k]; ro2 = h @ root2 + b2 ------------------
    {
        const int tiles_n = F_O / 16;                   // 1
        const int tiles_total = (NN / 16) * tiles_n;    // 1250
        const int grid = (tiles_total + 7) / 8;
        for (int k = 0; k < KT; ++k)
            wmma_gemm_f32<<<grid, BS, 0, stream>>>(
                h, F_H, W2 + (size_t)k * F_H * F_O, F_O, nullptr,
                xw2 + (size_t)k * NN * F_O, F_O, F_H, tiles_n, tiles_total);
        wmma_gemm_f32<<<grid, BS, 0, stream>>>(
            h, F_H, root2, F_O, b2, ro2, F_O, F_H, tiles_n, tiles_total);
    }

    // Edge blend + scatter (F_O=16 -> 4 float4 chunks/edge), reuse cnt.
    {
        const int total = EE << 2;
        edge_scatter<<<(total + BS - 1) / BS, BS, 0, stream>>>(
            xw2, eidx, eattr, agg2, nullptr, F_O, 2);
    }

    // mean + root + relu + log_softmax -> out
    finalize_logsoftmax<<<(NN + BS - 1) / BS, BS, 0, stream>>>(agg2, cnt, ro2, out);
}